// NodeModelBase_21947282882707
// MI455X (gfx1250) — compile-verified
//
#include <hip/hip_runtime.h>
#include <hip/hip_bf16.h>

typedef __attribute__((ext_vector_type(2))) float v2f;
typedef __attribute__((ext_vector_type(8))) float v8f;

// ---------------------------------------------------------------------------
// Kernel 0: zero the output accumulator (N*C) and the degree buffer (N).
// Must run every launch: harness does not re-poison between replays and we
// accumulate with atomics.
// ---------------------------------------------------------------------------
__global__ void zero_kernel(float* __restrict__ out, float* __restrict__ deg,
                            int n_out, int n_deg) {
    int i = blockIdx.x * blockDim.x + threadIdx.x;
    if (i < n_out) out[i] = 0.0f;
    if (i < n_deg) deg[i] = 0.0f;
}

// ---------------------------------------------------------------------------
// Kernel 1: deg[src[e]] += 1  (out-degree of source nodes)
// ---------------------------------------------------------------------------
__global__ void degree_kernel(const int* __restrict__ ei, float* __restrict__ deg,
                              int n_edges) {
    int e = blockIdx.x * blockDim.x + threadIdx.x;
    if (e < n_edges) {
        atomicAdd(&deg[ei[e]], 1.0f);
    }
}

// ---------------------------------------------------------------------------
// Kernel 2: dinv[n] = deg[n] > 0 ? deg[n]^-1/2 : 0
// ---------------------------------------------------------------------------
__global__ void dinv_kernel(const float* __restrict__ deg, float* __restrict__ dinv,
                            int n_nodes) {
    int n = blockIdx.x * blockDim.x + threadIdx.x;
    if (n < n_nodes) {
        float d = deg[n];
        dinv[n] = (d > 0.0f) ? rsqrtf(d) : 0.0f;
    }
}

// ---------------------------------------------------------------------------
// Kernel 3: node projection via V_WMMA_F32_16X16X4_F32.
//   s[n] = x[n,:] . w_src ;  t[n] = x[n,:] . w_tgt
// Each wave32 owns a 16-row tile of x (16 nodes x 32 features) and runs 8
// chained K=4 WMMAs against B with w_src in column N=0 and w_tgt in N=1.
//
// ISA layouts (CDNA5 ISA):
//   A (16x4 f32): lanes 0-15 -> M=lane, v0=K0 v1=K1 ; lanes 16-31 -> M=lane-16,
//                 v0=K2 v1=K3.
//   B (4x16 f32): mirrored -> N=lane%16, lanes 0-15 hold K0/K1, 16-31 K2/K3.
//   D (16x16 f32): lane -> N=lane%16 ; vgpr v -> M=v (lanes 0-15) / M=v+8.
//
// All A/B fragments are loaded unconditionally (branch-free, 8B-aligned b64
// loads) and selected with v_cndmask, so EXEC stays all-1s with no per-step
// exec save/restore, and the 8 WMMAs issue back-to-back on register operands.
// ---------------------------------------------------------------------------
__global__ void proj_wmma_kernel(const float* __restrict__ x,
                                 const float* __restrict__ w_src,
                                 const float* __restrict__ w_tgt,
                                 float* __restrict__ s, float* __restrict__ t,
                                 int n_nodes) {
    const int lane = threadIdx.x & 31;
    const int wave = (blockIdx.x * blockDim.x + threadIdx.x) >> 5;
    const int row0 = wave << 4;            // first node of this wave's tile
    if (row0 >= n_nodes) return;           // wave-uniform exit: EXEC stays full

    const int  m    = lane & 15;
    const bool hi   = lane >= 16;
    const int  koff = hi ? 2 : 0;
    const int  row  = row0 + m;
    const int  rowc = (row < n_nodes) ? row : (n_nodes - 1);
    const float valid = (row < n_nodes) ? 1.0f : 0.0f;
    const float* __restrict__ xrow = x + (size_t)rowc * 32 + koff;

    // Hoisted, branch-free fragment setup. k = kc*4 + koff is always even,
    // so every v2f load is 8-byte aligned.
    v2f afrag[8], bfrag[8];
#pragma unroll
    for (int kc = 0; kc < 8; ++kc) {
        const v2f a = *(const v2f*)(xrow + kc * 4);
        afrag[kc].x = a.x * valid;         // zero-pad tail rows
        afrag[kc].y = a.y * valid;
        const int k = kc * 4 + koff;
        const v2f ws = *(const v2f*)(w_src + k);
        const v2f wt = *(const v2f*)(w_tgt + k);
        bfrag[kc].x = (m == 0) ? ws.x : ((m == 1) ? wt.x : 0.0f);
        bfrag[kc].y = (m == 0) ? ws.y : ((m == 1) ? wt.y : 0.0f);
    }

    v8f acc = {};
#pragma unroll
    for (int kc = 0; kc < 8; ++kc) {
        // (neg_a, A, neg_b, B, c_mod, C, reuse_a, reuse_b)
        acc = __builtin_amdgcn_wmma_f32_16x16x4_f32(
            false, afrag[kc], false, bfrag[kc], (short)0, acc, false, false);
    }

    // Column 0 (s) lives in lanes 0 (M=0..7) and 16 (M=8..15);
    // column 1 (t) in lanes 1 and 17. Each of those 4 lanes writes 8 floats.
    const int mbase = row0 + (hi ? 8 : 0);
    float* __restrict__ dstp = (m == 0) ? s : ((m == 1) ? t : nullptr);
    if (dstp) {
#pragma unroll
        for (int v = 0; v < 8; ++v) {
            const int r = mbase + v;
            if (r < n_nodes) dstp[r] = acc[v];
        }
    }
}

// ---------------------------------------------------------------------------
// Kernel 4: gated, normalized scatter-add.
// Each wave owns 32 edges: every lane computes one edge's coefficient
//   coef = dinv[src]*dinv[dst] * sigmoid(s[src] + t[dst] + bias)
// then the wave sweeps the 32 edges; for edge j the coef/src/dst are
// shfl-broadcast and all 32 lanes do a coalesced 128B gather of x[src] and a
// coalesced 128B global_atomic_add_f32 into out[dst].
// ---------------------------------------------------------------------------
__global__ void edge_kernel(const int* __restrict__ ei,
                            const float* __restrict__ x,
                            const float* __restrict__ s,
                            const float* __restrict__ t,
                            const float* __restrict__ dinv,
                            const float* __restrict__ gate_bias,
                            float* __restrict__ out, int n_edges) {
    const int lane = threadIdx.x & 31;
    const long long wave = ((long long)blockIdx.x * blockDim.x + threadIdx.x) >> 5;
    const long long e = wave * 32 + lane;

    int   sIdx = 0, dIdx = 0;
    float coef = 0.0f;
    if (e < n_edges) {
        sIdx = ei[e];                       // row 0: source
        dIdx = ei[(size_t)n_edges + e];     // row 1: target
        float g = s[sIdx] + t[dIdx] + gate_bias[0];
        g = 1.0f / (1.0f + __expf(-g));     // sigmoid
        coef = dinv[sIdx] * dinv[dIdx] * g;
    }

#pragma unroll 4
    for (int j = 0; j < 32; ++j) {
        const float c  = __shfl(coef, j, 32);
        const int   sj = __shfl(sIdx, j, 32);
        const int   dj = __shfl(dIdx, j, 32);
        if (c != 0.0f) {                    // also skips padded tail edges
            const float v = x[(size_t)sj * 32 + lane];
            atomicAdd(&out[(size_t)dj * 32 + lane], c * v);
        }
    }
}

// ---------------------------------------------------------------------------
extern "C" void kernel_launch(void* const* d_in, const int* in_sizes, int n_in,
                              void* d_out, int out_size, void* d_ws, size_t ws_size,
                              hipStream_t stream) {
    const float* x         = (const float*)d_in[0];   // (N, C)
    const int*   ei        = (const int*)  d_in[1];   // (2, E)
    const float* w_src     = (const float*)d_in[2];   // (C,)
    const float* w_tgt     = (const float*)d_in[3];   // (C,)
    const float* gate_bias = (const float*)d_in[4];   // (1,)
    float* out = (float*)d_out;

    const int C  = in_sizes[2];            // 32
    const int NC = in_sizes[0];            // N*C
    const int N  = NC / C;                 // 100000
    const int E  = in_sizes[1] / 2;        // 2000000

    // Workspace layout: deg | dinv | s | t  (4 * N floats = 1.6 MB)
    float* deg  = (float*)d_ws;
    float* dinv = deg  + N;
    float* sbuf = dinv + N;
    float* tbuf = sbuf + N;

    const int T = 256;                     // 8 wave32s per block

    // 0: zero accumulators (every call: atomics accumulate)
    {
        int zmax = (NC > N) ? NC : N;
        zero_kernel<<<(zmax + T - 1) / T, T, 0, stream>>>(out, deg, NC, N);
    }
    // 1: out-degree of sources
    degree_kernel<<<(E + T - 1) / T, T, 0, stream>>>(ei, deg, E);
    // 2: deg^-1/2 with 0 for isolated nodes
    dinv_kernel<<<(N + T - 1) / T, T, 0, stream>>>(deg, dinv, N);
    // 3: WMMA node projections s, t
    {
        int nwaves   = (N + 15) / 16;
        int nthreads = nwaves * 32;
        proj_wmma_kernel<<<(nthreads + T - 1) / T, T, 0, stream>>>(
            x, w_src, w_tgt, sbuf, tbuf, N);
    }
    // 4: gated normalized scatter-add
    {
        long long ewaves   = ((long long)E + 31) / 32;
        long long ethreads = ewaves * 32;
        edge_kernel<<<(unsigned)((ethreads + T - 1) / T), T, 0, stream>>>(
            ei, x, sbuf, tbuf, dinv, gate_bias, out, E);
    }
}